// CascadedABCDCircuit_50783693308371
// MI455X (gfx1250) — compile-verified
//
#include <hip/hip_runtime.h>

#define VEC 4      // frequencies per thread (two v2f groups)
#define NOPS 12

typedef float v4f __attribute__((ext_vector_type(4)));
typedef float v2f __attribute__((ext_vector_type(2)));

// Packed complex mul-add over 2 frequency lanes (SoA):
//   (ar + i*ai) += (xr + i*xi) * (mr + i*mi), elementwise per lane.
// 4x v_pk_fma_f32 (8 scalar FMAs) per call; fneg folds into VOP3P NEG bits.
__device__ __forceinline__ void pk_cmadd(v2f& ar, v2f& ai,
                                         v2f xr, v2f xi,
                                         v2f mr, v2f mi) {
    ar = __builtin_elementwise_fma(xr, mr, ar);
    ar = __builtin_elementwise_fma(-xi, mi, ar);
    ai = __builtin_elementwise_fma(xr, mi, ai);
    ai = __builtin_elementwise_fma(xi, mr, ai);
}

__global__ __launch_bounds__(256) void cascade_abcd_kernel(
    const float* __restrict__ values,   // [B, 12]
    const float* __restrict__ freq,     // [F]
    float* __restrict__ out,            // [8, B, F]
    int B, int F)
{
    constexpr int OPS[NOPS] = {3, 0, 1, 2, 3, 0, 1, 2, 3, 0, 1, 2};
    constexpr float TWO_PI = 6.28318530717958647692f;

    const v2f EPS2 = {1e-30f, 1e-30f};
    const v2f QLI  = {1.0f / 50.0f,  1.0f / 50.0f};
    const v2f QCI  = {1.0f / 100.0f, 1.0f / 100.0f};

    const int b  = blockIdx.y;
    const int f0 = (blockIdx.x * blockDim.x + threadIdx.x) * VEC;

    __shared__ float s_vals[NOPS];
    const float* vrow = values + (size_t)b * NOPS;

#if defined(__gfx1250__) && __has_builtin(__builtin_amdgcn_global_load_async_to_lds_b32)
    // CDNA5 async global->LDS copy of the 12 row values (ASYNCcnt tracked).
    if (threadIdx.x < NOPS) {
        typedef __attribute__((address_space(1))) int* gptr_t;
        typedef __attribute__((address_space(3))) int* lptr_t;
        int* gsrc = (int*)(vrow + threadIdx.x);
        int* ldst = (int*)(s_vals + threadIdx.x);
        __builtin_amdgcn_global_load_async_to_lds_b32(
            (gptr_t)gsrc, (lptr_t)ldst, /*imm_offset=*/0, /*cpol=*/0);
    }
#if __has_builtin(__builtin_amdgcn_s_wait_asynccnt)
    __builtin_amdgcn_s_wait_asynccnt(0);
#else
    asm volatile("s_wait_asynccnt 0" ::: "memory");
#endif
#else
    if (threadIdx.x < NOPS) s_vals[threadIdx.x] = vrow[threadIdx.x];
#endif
    __syncthreads();

    if (f0 + VEC > F) return;

    float v[NOPS];
#pragma unroll
    for (int i = 0; i < NOPS; ++i) v[i] = s_vals[i];

    const v4f fr = *reinterpret_cast<const v4f*>(freq + f0);
    v2f omp[2];
    omp[0] = (v2f){TWO_PI * fr.x, TWO_PI * fr.y};
    omp[1] = (v2f){TWO_PI * fr.z, TWO_PI * fr.w};

    // SoA complex state, 2 freq lanes per v2f, 2 groups.
    v2f Are[2], Aim[2], Bre[2], Bim[2], Cre[2], Cim[2], Dre[2], Dim[2];
#pragma unroll
    for (int g = 0; g < 2; ++g) {
        Are[g] = (v2f){1.f, 1.f}; Aim[g] = (v2f){0.f, 0.f};
        Bre[g] = (v2f){0.f, 0.f}; Bim[g] = (v2f){0.f, 0.f};
        Cre[g] = (v2f){0.f, 0.f}; Cim[g] = (v2f){0.f, 0.f};
        Dre[g] = (v2f){1.f, 1.f}; Dim[g] = (v2f){0.f, 0.f};
    }

#pragma unroll
    for (int i = 0; i < NOPS; ++i) {
        const int code = OPS[i];
#pragma unroll
        for (int g = 0; g < 2; ++g) {
            const v2f wv = omp[g] * v[i];                       // v_pk_mul_f32
            if (code == 0) {            // SERIES_L: Z = wv/Q_L + i*wv
                const v2f zr = wv * QLI;
                pk_cmadd(Bre[g], Bim[g], Are[g], Aim[g], zr, wv);
                pk_cmadd(Dre[g], Dim[g], Cre[g], Cim[g], zr, wv);
            } else if (code == 1) {     // SERIES_C: Z = 1/(Y+eps), Y = wv/Q_C + i*wv
                const v2f yr  = __builtin_elementwise_fma(wv, QCI, EPS2);
                const v2f den = __builtin_elementwise_fma(yr, yr, wv * wv);
                v2f r;
                r.x = __builtin_amdgcn_rcpf(den.x);             // v_rcp_f32
                r.y = __builtin_amdgcn_rcpf(den.y);
                const v2f zr = yr * r;
                const v2f zi = -(wv * r);
                pk_cmadd(Bre[g], Bim[g], Are[g], Aim[g], zr, zi);
                pk_cmadd(Dre[g], Dim[g], Cre[g], Cim[g], zr, zi);
            } else if (code == 2) {     // SHUNT_L: Y = 1/(Z+eps), Z = wv/Q_L + i*wv
                const v2f zr  = __builtin_elementwise_fma(wv, QLI, EPS2);
                const v2f den = __builtin_elementwise_fma(zr, zr, wv * wv);
                v2f r;
                r.x = __builtin_amdgcn_rcpf(den.x);
                r.y = __builtin_amdgcn_rcpf(den.y);
                const v2f yr = zr * r;
                const v2f yi = -(wv * r);
                pk_cmadd(Are[g], Aim[g], Bre[g], Bim[g], yr, yi);
                pk_cmadd(Cre[g], Cim[g], Dre[g], Dim[g], yr, yi);
            } else {                    // SHUNT_C: Y = wv/Q_C + i*wv
                const v2f yr = wv * QCI;
                pk_cmadd(Are[g], Aim[g], Bre[g], Bim[g], yr, wv);
                pk_cmadd(Cre[g], Cim[g], Dre[g], Dim[g], yr, wv);
            }
        }
    }

    // Output layout: [k=0..7][b][f], k order = A.re A.im B.re B.im C.re C.im D.re D.im
    const size_t plane = (size_t)B * (size_t)F;
    float* o = out + (size_t)b * (size_t)F + (size_t)f0;

    v4f t;
    t = (v4f){Are[0].x, Are[0].y, Are[1].x, Are[1].y};
    __builtin_nontemporal_store(t, reinterpret_cast<v4f*>(o + 0 * plane));
    t = (v4f){Aim[0].x, Aim[0].y, Aim[1].x, Aim[1].y};
    __builtin_nontemporal_store(t, reinterpret_cast<v4f*>(o + 1 * plane));
    t = (v4f){Bre[0].x, Bre[0].y, Bre[1].x, Bre[1].y};
    __builtin_nontemporal_store(t, reinterpret_cast<v4f*>(o + 2 * plane));
    t = (v4f){Bim[0].x, Bim[0].y, Bim[1].x, Bim[1].y};
    __builtin_nontemporal_store(t, reinterpret_cast<v4f*>(o + 3 * plane));
    t = (v4f){Cre[0].x, Cre[0].y, Cre[1].x, Cre[1].y};
    __builtin_nontemporal_store(t, reinterpret_cast<v4f*>(o + 4 * plane));
    t = (v4f){Cim[0].x, Cim[0].y, Cim[1].x, Cim[1].y};
    __builtin_nontemporal_store(t, reinterpret_cast<v4f*>(o + 5 * plane));
    t = (v4f){Dre[0].x, Dre[0].y, Dre[1].x, Dre[1].y};
    __builtin_nontemporal_store(t, reinterpret_cast<v4f*>(o + 6 * plane));
    t = (v4f){Dim[0].x, Dim[0].y, Dim[1].x, Dim[1].y};
    __builtin_nontemporal_store(t, reinterpret_cast<v4f*>(o + 7 * plane));
}

extern "C" void kernel_launch(void* const* d_in, const int* in_sizes, int n_in,
                              void* d_out, int out_size, void* d_ws, size_t ws_size,
                              hipStream_t stream) {
    const float* values = (const float*)d_in[0];   // (1024, 12) f32
    const float* freq   = (const float*)d_in[1];   // (8192,)    f32
    float* out          = (float*)d_out;           // (8, 1024, 8192) f32

    const int B = in_sizes[0] / NOPS;
    const int F = in_sizes[1];

    dim3 block(256);
    dim3 grid((unsigned)((F + 256 * VEC - 1) / (256 * VEC)), (unsigned)B);
    hipLaunchKernelGGL(cascade_abcd_kernel, grid, block, 0, stream,
                       values, freq, out, B, F);
}